// VQC_40939628265591
// MI455X (gfx1250) — compile-verified
//
#include <hip/hip_runtime.h>

typedef float v2f __attribute__((ext_vector_type(2)));
typedef float v8f __attribute__((ext_vector_type(8)));

struct cplx { float re, im; };

__device__ __forceinline__ cplx cmul(cplx a, cplx b) {
    cplx r; r.re = a.re * b.re - a.im * b.im; r.im = a.re * b.im + a.im * b.re; return r;
}

// CNOT ring permutation applied in circuit order: CNOT(0,1),CNOT(1,2),CNOT(2,3),CNOT(3,0)
// basis index b = p0*8 + p1*4 + p2*2 + p3 ; CNOT(c,t): p_t ^= p_c
__device__ __forceinline__ int cnot_ring_perm(int j) {
    int p0 = (j >> 3) & 1, p1 = (j >> 2) & 1, p2 = (j >> 1) & 1, p3 = j & 1;
    p1 ^= p0; p2 ^= p1; p3 ^= p2; p0 ^= p3;
    return (p0 << 3) | (p1 << 2) | (p2 << 1) | p3;
}

// 2x2 complex gate RZ(z)*RY(y)*RX(x)  (circuit applies RX, then RY, then RZ)
__device__ __forceinline__ void site_gate(float x, float y, float z, cplx g[2][2]) {
    float cx = __cosf(0.5f * x), sx = __sinf(0.5f * x);
    float cy = __cosf(0.5f * y), sy = __sinf(0.5f * y);
    float cz = __cosf(0.5f * z), sz = __sinf(0.5f * z);
    // RY*RX
    cplx m00 = { cy * cx,  sy * sx };
    cplx m01 = { -sy * cx, -cy * sx };
    cplx m10 = { sy * cx, -cy * sx };
    cplx m11 = { cy * cx, -sy * sx };
    cplx e0 = { cz, -sz };   // exp(-i z/2)
    cplx e1 = { cz,  sz };   // exp(+i z/2)
    g[0][0] = cmul(e0, m00); g[0][1] = cmul(e0, m01);
    g[1][0] = cmul(e1, m10); g[1][1] = cmul(e1, m11);
}

// Build 16x16 = kron of four 2x2 complex matrices into LDS (8 entries per lane).
__device__ __forceinline__ void build_kron(const cplx g[4][2][2],
                                           float (*Mr)[16], float (*Mi)[16], int lane) {
    for (int t = 0; t < 8; ++t) {
        int e = lane * 8 + t;
        int bo = e >> 4, bi = e & 15;
        cplx v = { 1.0f, 0.0f };
        for (int n = 0; n < 4; ++n) {
            int po = (bo >> (3 - n)) & 1;
            int pi = (bi >> (3 - n)) & 1;
            v = cmul(v, g[n][po][pi]);
        }
        Mr[bo][bi] = v.re;
        Mi[bo][bi] = v.im;
    }
}

// ---- WMMA f32 16x16x4 fragment helpers (wave32 ISA layouts) ----
// A 16x4: lanes 0-15 row M=lane hold K=k0,k0+1 in v0,v1; lanes 16-31 hold K=k0+2,k0+3.
__device__ __forceinline__ v2f load_A(const float (*M)[16], int lane, int k0, float sgn) {
    int row = lane & 15;
    int kk  = k0 + ((lane >> 4) << 1);
    v2f a; a.x = sgn * M[row][kk]; a.y = sgn * M[row][kk + 1];
    return a;
}
// B 4x16: lanes 0-15 col N=lane hold K=k0,k0+1; lanes 16-31 hold K=k0+2,k0+3.
__device__ __forceinline__ v2f load_B(const float (*M)[16], int lane, int k0) {
    int col = lane & 15;
    int kk  = k0 + ((lane >> 4) << 1);
    v2f b; b.x = M[kk][col]; b.y = M[kk + 1][col];
    return b;
}
// C/D 16x16: VGPR r: lanes 0-15 -> (M=r, N=lane); lanes 16-31 -> (M=r+8, N=lane-16).
__device__ __forceinline__ void store_D(float (*M)[16], int lane, v8f d) {
    int col   = lane & 15;
    int rbase = (lane >> 4) << 3;
    for (int r = 0; r < 8; ++r) M[rbase + r][col] = d[r];
}

__global__ __launch_bounds__(32)
void VQC_40939628265591_kernel(const float* __restrict__ psi,
                               const float* __restrict__ ring_init,
                               const float* __restrict__ pRX,
                               const float* __restrict__ pRY,
                               const float* __restrict__ pRZ,
                               float* __restrict__ out) {
    __shared__ float Ur[16][16], Ui[16][16];   // running circuit unitary (complex)
    __shared__ float Tr[16][16], Ti[16][16];   // CNOT-permuted copy
    __shared__ float Gr[16][16], Gi[16][16];   // per-layer gate kron matrix

    const int lane = threadIdx.x;              // exactly one wave32

    // ---- data-encoding layer: U = kron_n RX(psi[n]) ----
    {
        cplx g[4][2][2];
        for (int n = 0; n < 4; ++n) {
            float c = __cosf(0.5f * psi[n]);
            float s = __sinf(0.5f * psi[n]);
            g[n][0][0] = { c, 0.0f };  g[n][0][1] = { 0.0f, -s };
            g[n][1][0] = { 0.0f, -s }; g[n][1][1] = { c, 0.0f };
        }
        build_kron(g, Ur, Ui, lane);
    }
    __syncthreads();

    // ---- 10 variational layers ----
    for (int d = 0; d < 10; ++d) {
        // CNOT ring = basis permutation: T = P * U  (row permute, zero flops)
        for (int t = 0; t < 8; ++t) {
            int e = lane * 8 + t;
            int row = e >> 4, col = e & 15;
            int srow = cnot_ring_perm(row);
            Tr[srow][col] = Ur[row][col];
            Ti[srow][col] = Ui[row][col];
        }
        // gate layer matrix G = kron_n (RZ*RY*RX)
        cplx g[4][2][2];
        for (int n = 0; n < 4; ++n)
            site_gate(pRX[d * 4 + n], pRY[d * 4 + n], pRZ[d * 4 + n], g[n]);
        __syncthreads();
        build_kron(g, Gr, Gi, lane);
        __syncthreads();

        // complex 16x16 update U <- G * T via v_wmma_f32_16x16x4_f32 (K chained 4x):
        //   Ur' = Gr*Tr - Gi*Ti ;  Ui' = Gr*Ti + Gi*Tr
        v8f accr = {};
        v8f acci = {};
        for (int k0 = 0; k0 < 16; k0 += 4) {
            v2f agr  = load_A(Gr, lane, k0,  1.0f);
            v2f agi  = load_A(Gi, lane, k0,  1.0f);
            v2f agin = load_A(Gi, lane, k0, -1.0f);  // f32 WMMA NEG only allows C-neg, so negate A in VALU
            v2f btr  = load_B(Tr, lane, k0);
            v2f bti  = load_B(Ti, lane, k0);
            accr = __builtin_amdgcn_wmma_f32_16x16x4_f32(false, agr,  false, btr, (short)0, accr, false, false);
            accr = __builtin_amdgcn_wmma_f32_16x16x4_f32(false, agin, false, bti, (short)0, accr, false, false);
            acci = __builtin_amdgcn_wmma_f32_16x16x4_f32(false, agr,  false, bti, (short)0, acci, false, false);
            acci = __builtin_amdgcn_wmma_f32_16x16x4_f32(false, agi,  false, btr, (short)0, acci, false, false);
        }
        store_D(Ur, lane, accr);
        store_D(Ui, lane, acci);
        __syncthreads();
    }

    // ---- final state k = U * s0, s0[j] = prod_n ring_init[n][p_n] (real) ----
    int i = lane & 15;   // duplicated across lane halves; keeps indexing in range
    float kr = 0.0f, ki = 0.0f;
    for (int j = 0; j < 16; ++j) {
        float s0 = ring_init[0 * 2 + ((j >> 3) & 1)] *
                   ring_init[1 * 2 + ((j >> 2) & 1)] *
                   ring_init[2 * 2 + ((j >> 1) & 1)] *
                   ring_init[3 * 2 + (j & 1)];
        kr += Ur[i][j] * s0;
        ki += Ui[i][j] * s0;
    }
    float amp2 = kr * kr + ki * ki;
    // outputs: |k[0,0,0,0]|^2 -> b=0 ; |k[0,0,1,0]|^2 -> b=2 ; |k[1,1,1,1]|^2 -> b=15
    if (lane == 0)  out[0] = amp2;
    if (lane == 2)  out[1] = amp2;
    if (lane == 15) out[2] = amp2;
}

extern "C" void kernel_launch(void* const* d_in, const int* in_sizes, int n_in,
                              void* d_out, int out_size, void* d_ws, size_t ws_size,
                              hipStream_t stream) {
    const float* psi       = (const float*)d_in[0];  // (4,)
    const float* ring_init = (const float*)d_in[1];  // (4,2)
    const float* pRX       = (const float*)d_in[2];  // (10,4)
    const float* pRY       = (const float*)d_in[3];  // (10,4)
    const float* pRZ       = (const float*)d_in[4];  // (10,4)
    float* out = (float*)d_out;                      // (1,3) float32
    (void)in_sizes; (void)n_in; (void)out_size; (void)d_ws; (void)ws_size;

    VQC_40939628265591_kernel<<<1, 32, 0, stream>>>(psi, ring_init, pRX, pRY, pRZ, out);
}